// TransD_45148696216013
// MI455X (gfx1250) — compile-verified
//
#include <hip/hip_runtime.h>

typedef __attribute__((ext_vector_type(2))) float v2f;
typedef __attribute__((ext_vector_type(4))) float v4f;
typedef __attribute__((ext_vector_type(8))) float v8f;

#define EMB_D 128

// One wave (32 lanes) processes 16 triples.
//  Phase 1: V_WMMA_F32_16X16X4_F32, K-loop over 128 in steps of 4.
//           A = hp-tile (16x4 f32), B = h-tile (4x16 f32, B[k][n] = h[n][k]).
//           C[m][n] = sum_k hp[m][k]*h[n][k]; we keep the diagonal (the 16 dots).
//           Both A and B per the ISA f32 layouts: lane l supplies row (l&15),
//           K elements 2*(l>>4)+{0,1}  -> one 8-byte load per lane per matrix.
//  Phase 2: wave-coalesced float4 epilogue: out = h - t + r + rp*(sh - st).
__global__ __launch_bounds__(256) void transd_kernel(
    const int* __restrict__ head,
    const int* __restrict__ relation,
    const int* __restrict__ tail,
    const float* __restrict__ ent_emb,
    const float* __restrict__ ent_map_emb,
    const float* __restrict__ rel_emb,
    const float* __restrict__ rel_map_emb,
    float* __restrict__ out)
{
    const int lane  = threadIdx.x & 31;
    const int wave  = threadIdx.x >> 5;
    const long base = ((long)blockIdx.x * 8 + wave) * 16;   // first triple of this wave
    const int mrow  = lane & 15;                            // which of the 16 triples this lane serves
    const int khalf = lane >> 4;                            // K sub-pair (0 -> K{0,1}, 1 -> K{2,3})

    const int hidx = head    [base + mrow];
    const int tidx = tail    [base + mrow];
    const int ridx = relation[base + mrow];

    const float* __restrict__ hrow  = ent_emb     + (long)hidx * EMB_D;
    const float* __restrict__ hprow = ent_map_emb + (long)hidx * EMB_D;
    const float* __restrict__ trow  = ent_emb     + (long)tidx * EMB_D;
    const float* __restrict__ tprow = ent_map_emb + (long)tidx * EMB_D;

    v8f ch = {0.f,0.f,0.f,0.f,0.f,0.f,0.f,0.f};
    v8f ct = {0.f,0.f,0.f,0.f,0.f,0.f,0.f,0.f};
    const int coloff = khalf * 2;

    // ---- Phase 1: batched dot products via WMMA (f32 in, f32 accumulate) ----
#pragma unroll
    for (int k = 0; k < EMB_D / 4; ++k) {
        const int c = 4 * k + coloff;
        v2f a_h = *(const v2f*)(hprow + c);     // A[m][k'] = hp[m][k']
        v2f b_h = *(const v2f*)(hrow  + c);     // B[k'][n] = h[n][k']  (same lane pattern)
        ch = __builtin_amdgcn_wmma_f32_16x16x4_f32(
                 false, a_h, false, b_h, (short)0, ch, false, false);
        v2f a_t = *(const v2f*)(tprow + c);
        v2f b_t = *(const v2f*)(trow  + c);
        ct = __builtin_amdgcn_wmma_f32_16x16x4_f32(
                 false, a_t, false, b_t, (short)0, ct, false, false);
    }

    // diff of the two accumulators; diagonal holds (hp.h - tp.t) per triple
    v8f e = ch - ct;

    // ---- Phase 2: coalesced epilogue, one 512B row per wave iteration ----
#pragma unroll
    for (int m = 0; m < 16; ++m) {
        // C/D layout: m<8 -> VGPR m, lane m ; m>=8 -> VGPR m-8, lane m+16
        float sd;
        if (m < 8) sd = __shfl(e[m],     m,      32);
        else       sd = __shfl(e[m - 8], m + 16, 32);
        const int hm = __shfl(hidx, m, 32);
        const int tm = __shfl(tidx, m, 32);
        const int rm = __shfl(ridx, m, 32);

        const v4f hv  = *(const v4f*)(ent_emb     + (long)hm * EMB_D + 4 * lane);
        const v4f tv  = *(const v4f*)(ent_emb     + (long)tm * EMB_D + 4 * lane);
        const v4f rv  = *(const v4f*)(rel_emb     + (long)rm * EMB_D + 4 * lane);
        const v4f rpv = *(const v4f*)(rel_map_emb + (long)rm * EMB_D + 4 * lane);

        v4f o = hv - tv + rv + rpv * sd;
        *(v4f*)(out + (base + m) * EMB_D + 4 * lane) = o;
    }
}

extern "C" void kernel_launch(void* const* d_in, const int* in_sizes, int n_in,
                              void* d_out, int out_size, void* d_ws, size_t ws_size,
                              hipStream_t stream) {
    const int*   head        = (const int*)  d_in[0];
    const int*   relation    = (const int*)  d_in[1];
    const int*   tail        = (const int*)  d_in[2];
    const float* ent_emb     = (const float*)d_in[3];
    const float* ent_map_emb = (const float*)d_in[4];
    const float* rel_emb     = (const float*)d_in[5];
    const float* rel_map_emb = (const float*)d_in[6];
    float* out = (float*)d_out;

    const int B = in_sizes[0];        // 262144; divisible by 128
    const int blocks = B / 128;       // 8 waves/block * 16 triples/wave

    transd_kernel<<<blocks, 256, 0, stream>>>(
        head, relation, tail, ent_emb, ent_map_emb, rel_emb, rel_map_emb, out);
}